// Embedding_1314259993177
// MI455X (gfx1250) — compile-verified
//
#include <hip/hip_runtime.h>
#include <cstdint>

// Embedding lookup: out[t, :] = W[:, e[t]] + b
//   e: int32 [8192], W: f32 [128, 32000] (row-major), b: f32 [128]
//   out: f32 [8192, 128]
//
// Memory-bound gather (W is 16 MB -> L2-resident on MI455X's 192 MB L2).
// One wave32 per token; lane l produces out elements 4l..4l+3 (float4 store,
// coalesced 512B per wave). Bias is staged into LDS once per block via the
// CDNA5 async-to-LDS path (global_load_async_to_lds_b128 + s_wait_asynccnt).

#define VOCAB 32000
#define EMBED 128
#define NTOK  (4 * 2048)          // 8192 tokens
#define TOK_PER_BLOCK 8           // 8 waves of 32 per block

__global__ __launch_bounds__(256) void embed_gather_kernel(
    const int* __restrict__ e,
    const float* __restrict__ W,
    const float* __restrict__ bias,
    float* __restrict__ out) {
  __shared__ float s_bias[EMBED];

  const int tid  = threadIdx.x;
  const int wave = tid >> 5;
  const int lane = tid & 31;

  // ---- CDNA5 async copy: wave 0 stages the full 512B bias into LDS with a
  // single global_load_async_to_lds_b128 (each lane moves 16B). Tracked by
  // ASYNCcnt, completed with s_wait_asynccnt, published via the barrier. ----
  if (wave == 0) {
    uint64_t ga = (uint64_t)(uintptr_t)bias + (uint64_t)lane * 16u;
    uint32_t la = (uint32_t)(uintptr_t)s_bias + (uint32_t)lane * 16u;
    asm volatile("global_load_async_to_lds_b128 %0, %1, off"
                 :: "v"(la), "v"(ga)
                 : "memory");
    asm volatile("s_wait_asynccnt 0" ::: "memory");
  }

  // ---- Gather: one wave per token. Token id is wave-uniform; force it into
  // an SGPR so the strided W loads lower to saddr-form global loads. ----
  const int tok_idx = blockIdx.x * TOK_PER_BLOCK + wave;
  const int tok = __builtin_amdgcn_readfirstlane(e[tok_idx]);

  const float* col = W + tok;           // column e[t] of W, stride VOCAB
  const size_t k0 = (size_t)lane * 4;
  float r0 = col[(k0 + 0) * VOCAB];
  float r1 = col[(k0 + 1) * VOCAB];
  float r2 = col[(k0 + 2) * VOCAB];
  float r3 = col[(k0 + 3) * VOCAB];

  __syncthreads();                      // bias now visible in LDS to all waves

  const float4 bb = *reinterpret_cast<const float4*>(&s_bias[k0]);
  float4 o;
  o.x = r0 + bb.x;
  o.y = r1 + bb.y;
  o.z = r2 + bb.z;
  o.w = r3 + bb.w;

  // Coalesced float4 store: wave writes one contiguous 512B row.
  reinterpret_cast<float4*>(out)[(size_t)tok_idx * (EMBED / 4) + lane] = o;
}

extern "C" void kernel_launch(void* const* d_in, const int* in_sizes, int n_in,
                              void* d_out, int out_size, void* d_ws, size_t ws_size,
                              hipStream_t stream) {
  (void)in_sizes; (void)n_in; (void)out_size; (void)d_ws; (void)ws_size;

  const int*   e = (const int*)d_in[0];    // [4, 2048] token ids
  const float* W = (const float*)d_in[1];  // [128, 32000]
  const float* b = (const float*)d_in[2];  // [128]
  float* out = (float*)d_out;              // [4, 2048, 128]

  dim3 grid(NTOK / TOK_PER_BLOCK);  // 1024 blocks
  dim3 block(256);                  // 8 wave32s
  embed_gather_kernel<<<grid, block, 0, stream>>>(e, W, b, out);
}